// BasicLSTM_78486232367305
// MI455X (gfx1250) — compile-verified
//
#include <hip/hip_runtime.h>
#include <hip/hip_bf16.h>

typedef __attribute__((ext_vector_type(2))) float v2f;
typedef __attribute__((ext_vector_type(8))) float v8f;

#define B_TOT 256
#define T_LEN 1024
#define D_IN  16
#define H_DIM 128
#define KDIM  144          // D_IN + H_DIM
#define GCOLS 512          // 4*H
#define ROWS  16           // batch rows per block
#define NW    8            // waves per block
#define AS    148          // padded A row stride (floats)

// W stored fragment-major: wBuf[kc][col][k'] ; kc=k/4, k'=k&3.
// A lane's B fragment {W[k+kb][c], W[k+kb+1][c]} is contiguous -> one ds_load_b64,
// banks (4c+kb)%64 cover all 64 banks across the wave -> conflict free, no padding.

// ---- LDS layout (float offsets) ----
#define W_OFFS    0
#define A_OFFS    (KDIM * GCOLS)              // 73728
#define BIAS_OFFS (A_OFFS + ROWS * AS)        // 76096
#define LNG_OFFS  (BIAS_OFFS + GCOLS)
#define LNB_OFFS  (LNG_OFFS + H_DIM)
#define WOUT_OFFS (LNB_OFFS + H_DIM)
#define PLN_OFFS  (WOUT_OFFS + H_DIM)         // [NW][ROWS][2]
#define POUT_OFFS (PLN_OFFS + NW * ROWS * 2)  // [NW][ROWS]
#define STAT_OFFS (POUT_OFFS + NW * ROWS)     // [ROWS][2] mu, rstd
#define PO_OFFS_L (STAT_OFFS + ROWS * 2)      // prev_out[16]
#define EN_OFFS_L (PO_OFFS_L + ROWS)          // energy[16]
#define PX_OFFS_L (EN_OFFS_L + ROWS)          // prev_x[16]
#define BO_OFFS_L (PX_OFFS_L + ROWS)          // b_out scalar
#define LDS_FLOATS (BO_OFFS_L + 1)            // ~77457 floats = 309828 B < 320KB

// ---- output buffer layout (flat concat, floats) ----
#define OUT_OFF  0
#define ENER_OFF (B_TOT * T_LEN)
#define HO_OFF   (2 * B_TOT * T_LEN)
#define CO_OFF   (HO_OFF + B_TOT * H_DIM)
#define PO_OFF   (CO_OFF + B_TOT * H_DIM)
#define E_OFF    (PO_OFF + B_TOT)
#define PX_OFF   (E_OFF + B_TOT)

// Branchless fast activations: single v_exp_f32 + v_rcp_f32 each, no control flow.
__device__ __forceinline__ float sigmoid_fast(float x) {
    return __builtin_amdgcn_rcpf(1.0f + __expf(-x));
}
__device__ __forceinline__ float tanh_fast(float x) {
    // e = exp(-2|x|) in (0,1]; tanh(|x|) = (1-e)/(1+e); restore sign bitwise.
    float e = __expf(-2.0f * __builtin_fabsf(x));
    float t = (1.0f - e) * __builtin_amdgcn_rcpf(1.0f + e);
    return __builtin_copysignf(t, x);
}

__global__ __launch_bounds__(256) void lstm_persistent_kernel(
    const float* __restrict__ x, const float* __restrict__ W_all,
    const float* __restrict__ b_all, const float* __restrict__ ln_g,
    const float* __restrict__ ln_b, const float* __restrict__ W_out,
    const float* __restrict__ b_out, float* __restrict__ out)
{
    extern __shared__ float lds[];
    const int tid   = threadIdx.x;
    const int lane  = tid & 31;
    const int wave  = tid >> 5;
    const int hi    = lane >> 4;        // half-wave select
    const int n16   = lane & 15;        // M index for A-frag, N index for B/C-frag
    const int kb    = hi * 2;           // K sub-offset per ISA 32-bit A layout
    const int bBase = blockIdx.x * ROWS;
    const int hcol  = wave * 16 + n16;  // hidden column owned by this lane

    // ---------------- one-time staging ----------------
    for (int i = tid; i < KDIM * GCOLS; i += 256) {
        int r = i >> 9, c = i & 511;
        // fragment-major: [kc][col][k']
        lds[W_OFFS + ((r >> 2) << 11) + (c << 2) + (r & 3)] = W_all[i];
    }
    for (int i = tid; i < GCOLS; i += 256) lds[BIAS_OFFS + i] = b_all[i];
    for (int i = tid; i < H_DIM; i += 256) {
        lds[LNG_OFFS + i]  = ln_g[i];
        lds[LNB_OFFS + i]  = ln_b[i];
        lds[WOUT_OFFS + i] = W_out[i];
    }
    for (int i = tid; i < ROWS * H_DIM; i += 256) {      // h0 = 0
        int m = i >> 7, c = i & 127;
        lds[A_OFFS + m * AS + D_IN + c] = 0.0f;
    }
    if (tid < ROWS) {
        lds[PO_OFFS_L + tid] = 0.0f;
        lds[EN_OFFS_L + tid] = 0.0f;
        lds[PX_OFFS_L + tid] = 0.0f;
    }
    if (tid == 0) lds[BO_OFFS_L] = b_out[0];
    v8f cReg = {};                                        // c0 = 0 (16x16 tile)
    __syncthreads();

    const int colI = wave * 16 + n16;
    const int colF = colI + 128;
    const int colG = colI + 256;
    const int colO = colI + 384;
    const int cI4  = (colI << 2) + kb;   // fragment offsets within a k-chunk slab
    const int cF4  = (colF << 2) + kb;
    const int cG4  = (colG << 2) + kb;
    const int cO4  = (colO << 2) + kb;
    const int aBase = A_OFFS + n16 * AS + kb;

    for (int t = 0; t < T_LEN; ++t) {
        // stage x_t (16 rows x 16 features) into A-buffer cols [0,16)
        {
            int m = tid >> 4, d = tid & 15;
            lds[A_OFFS + m * AS + d] =
                x[((size_t)(bBase + m) * T_LEN + t) * D_IN + d];
        }
        __syncthreads();

        // ---- gates = [x_t, h] @ W_all : four 16x16 tiles per wave, fp32 WMMA ----
        // Per chunk: 1 A-frag b64 + 4 B-frag b64 loads feeding 4 WMMAs, no repacking.
        v8f accI = {}, accF = {}, accG = {}, accO = {};
        #pragma unroll 4
        for (int k = 0; k < KDIM; k += 4) {
            v2f a = *(const v2f*)&lds[aBase + k];
            const int wb = W_OFFS + (k << 9);            // k/4 * 2048
            v2f bI = *(const v2f*)&lds[wb + cI4];
            v2f bF = *(const v2f*)&lds[wb + cF4];
            v2f bG = *(const v2f*)&lds[wb + cG4];
            v2f bO = *(const v2f*)&lds[wb + cO4];
            accI = __builtin_amdgcn_wmma_f32_16x16x4_f32(false, a, false, bI, (short)0, accI, false, false);
            accF = __builtin_amdgcn_wmma_f32_16x16x4_f32(false, a, false, bF, (short)0, accF, false, false);
            accG = __builtin_amdgcn_wmma_f32_16x16x4_f32(false, a, false, bG, (short)0, accG, false, false);
            accO = __builtin_amdgcn_wmma_f32_16x16x4_f32(false, a, false, bO, (short)0, accO, false, false);
        }

        // ---- gate nonlinearities + cell update (all in registers, branchless) ----
        const float bi = lds[BIAS_OFFS + colI];
        const float bf = lds[BIAS_OFFS + colF];
        const float bg = lds[BIAS_OFFS + colG];
        const float bo = lds[BIAS_OFFS + colO];
        v8f zv;
        #pragma unroll
        for (int v = 0; v < 8; ++v) {
            float ig = sigmoid_fast(accI[v] + bi);
            float fg = sigmoid_fast(accF[v] + bf);
            float gg = tanh_fast(accG[v] + bg);
            float og = sigmoid_fast(accO[v] + bo);
            float cn = fg * cReg[v] + ig * gg;
            cReg[v]  = cn;
            zv[v]    = og * tanh_fast(cn);          // LN input
        }

        // ---- LayerNorm row stats: half-wave shuffle reduce + cross-wave LDS ----
        #pragma unroll
        for (int v = 0; v < 8; ++v) {
            float s = zv[v], s2 = s * s;
            s += __shfl_xor(s, 1, 32);  s2 += __shfl_xor(s2, 1, 32);
            s += __shfl_xor(s, 2, 32);  s2 += __shfl_xor(s2, 2, 32);
            s += __shfl_xor(s, 4, 32);  s2 += __shfl_xor(s2, 4, 32);
            s += __shfl_xor(s, 8, 32);  s2 += __shfl_xor(s2, 8, 32);
            if (n16 == 0) {
                int m = v + 8 * hi;
                lds[PLN_OFFS + (wave * ROWS + m) * 2 + 0] = s;
                lds[PLN_OFFS + (wave * ROWS + m) * 2 + 1] = s2;
            }
        }
        __syncthreads();
        if (tid < ROWS) {
            float s = 0.0f, s2 = 0.0f;
            #pragma unroll
            for (int w = 0; w < NW; ++w) {
                s  += lds[PLN_OFFS + (w * ROWS + tid) * 2 + 0];
                s2 += lds[PLN_OFFS + (w * ROWS + tid) * 2 + 1];
            }
            float mu  = s * (1.0f / H_DIM);
            float var = s2 * (1.0f / H_DIM) - mu * mu;
            lds[STAT_OFFS + tid * 2 + 0] = mu;
            lds[STAT_OFFS + tid * 2 + 1] = rsqrtf(var + 1e-5f);
        }
        __syncthreads();

        // ---- normalize, write h into A-buffer, project to scalar out ----
        const float lg = lds[LNG_OFFS + hcol];
        const float lb = lds[LNB_OFFS + hcol];
        const float wo = lds[WOUT_OFFS + hcol];
        #pragma unroll
        for (int v = 0; v < 8; ++v) {
            int m = v + 8 * hi;
            float mu = lds[STAT_OFFS + m * 2 + 0];
            float rs = lds[STAT_OFFS + m * 2 + 1];
            float hv = (zv[v] - mu) * rs * lg + lb;
            lds[A_OFFS + m * AS + D_IN + hcol] = hv;   // next step's A-matrix
            float po = hv * wo;
            po += __shfl_xor(po, 1, 32);
            po += __shfl_xor(po, 2, 32);
            po += __shfl_xor(po, 4, 32);
            po += __shfl_xor(po, 8, 32);
            if (n16 == 0) lds[POUT_OFFS + wave * ROWS + m] = po;
        }
        __syncthreads();

        // ---- finalize out / energy, stream per-step scalars to HBM ----
        if (tid < ROWS) {
            float o = lds[BO_OFFS_L];
            #pragma unroll
            for (int w = 0; w < NW; ++w) o += lds[POUT_OFFS + w * ROWS + tid];
            float curx = lds[A_OFFS + tid * AS + 0];
            float en = lds[EN_OFFS_L + tid] +
                       (o + lds[PO_OFFS_L + tid]) * 0.5f *
                       (curx - lds[PX_OFFS_L + tid]);
            size_t gi = (size_t)(bBase + tid) * T_LEN + t;
            out[OUT_OFF + gi]  = o;
            out[ENER_OFF + gi] = en;
            lds[PO_OFFS_L + tid] = o;
            lds[EN_OFFS_L + tid] = en;
            lds[PX_OFFS_L + tid] = curx;
        }
        __syncthreads();
    }

    // ---------------- final carry writeout ----------------
    for (int i = tid; i < ROWS * H_DIM; i += 256) {
        int m = i >> 7, c = i & 127;
        out[HO_OFF + (size_t)(bBase + m) * H_DIM + c] = lds[A_OFFS + m * AS + D_IN + c];
    }
    #pragma unroll
    for (int v = 0; v < 8; ++v) {
        int m = v + 8 * hi;
        out[CO_OFF + (size_t)(bBase + m) * H_DIM + hcol] = cReg[v];
    }
    if (tid < ROWS) {
        out[PO_OFF + bBase + tid] = lds[PO_OFFS_L + tid];
        out[E_OFF  + bBase + tid] = lds[EN_OFFS_L + tid];
        out[PX_OFF + bBase + tid] = lds[PX_OFFS_L + tid];
    }
}

extern "C" void kernel_launch(void* const* d_in, const int* in_sizes, int n_in,
                              void* d_out, int out_size, void* d_ws, size_t ws_size,
                              hipStream_t stream) {
    (void)in_sizes; (void)n_in; (void)d_ws; (void)ws_size; (void)out_size;
    const float* x     = (const float*)d_in[0];
    const float* W_all = (const float*)d_in[1];
    const float* b_all = (const float*)d_in[2];
    const float* ln_g  = (const float*)d_in[3];
    const float* ln_b  = (const float*)d_in[4];
    const float* W_out = (const float*)d_in[5];
    const float* b_out = (const float*)d_in[6];
    float* out = (float*)d_out;

    const size_t shmem = (size_t)LDS_FLOATS * sizeof(float);  // ~310 KB, fits 320 KB WGP LDS
    hipLaunchKernelGGL(lstm_persistent_kernel,
                       dim3(B_TOT / ROWS), dim3(256), shmem, stream,
                       x, W_all, b_all, ln_g, ln_b, W_out, b_out, out);
}